// DGLGraphConv_13580686590542
// MI455X (gfx1250) — compile-verified
//
#include <hip/hip_runtime.h>

typedef __attribute__((ext_vector_type(2))) float v2f;
typedef __attribute__((ext_vector_type(8))) float v8f;
typedef int v4i __attribute__((vector_size(16)));

static constexpr int IN_FEATS  = 256;
static constexpr int HIDDEN    = 128;
static constexpr int OUT_FEATS = 64;

#if __has_builtin(__builtin_amdgcn_global_load_async_to_lds_b128) && \
    __has_builtin(__builtin_amdgcn_s_wait_asynccnt)
#define HAS_ASYNC_LDS 1
#else
#define HAS_ASYNC_LDS 0
#endif

// ---------------- elementwise helpers ----------------

__global__ void k_zero(float* __restrict__ p, long long n) {
    long long i = (long long)blockIdx.x * blockDim.x + threadIdx.x;
    if (i < n) p[i] = 0.0f;
}

__global__ void k_deg(const int* __restrict__ src, const int* __restrict__ dst,
                      float* __restrict__ deg_out, float* __restrict__ deg_in, int E) {
    int e = blockIdx.x * blockDim.x + threadIdx.x;
    if (e < E) {
        atomicAdd(&deg_out[src[e]], 1.0f);
        atomicAdd(&deg_in[dst[e]], 1.0f);
    }
}

__global__ void k_norm(float* __restrict__ d, int n) {
    int i = blockIdx.x * blockDim.x + threadIdx.x;
    if (i < n) {
        float v = d[i];
        v = (v < 1.0f) ? 1.0f : v;
        d[i] = rsqrtf(v);
    }
}

// ---------------- WMMA GEMM: Y[nrows,OC] = (X * rowscale) @ W ----------------
// One wave computes a 16x16 tile via V_WMMA_F32_16X16X4_F32 (fp32, exact).
// Block = (OC/16) waves covering 16 rows x OC cols. A tile staged in LDS via
// CDNA5 async global->LDS copies (ASYNCcnt / s_wait_asynccnt) when available;
// row-scaling is applied at fragment load time (per-lane scale register).

template <int K, int OC>
__global__ __launch_bounds__(32 * (OC / 16))
void k_gemm_wmma(const float* __restrict__ X, const float* __restrict__ W,
                 const float* __restrict__ rowscale, float* __restrict__ Y,
                 int nrows) {
    __shared__ float As[16 * K];

    const int row0 = blockIdx.x * 16;
    const int tid  = threadIdx.x;
    constexpr int NTHREADS = 32 * (OC / 16);
    constexpr int NV4 = (16 * K) / 4;  // 16-byte packets in the A tile

    // ---- stage 16 x K A tile into LDS (unscaled) ----
    for (int i = tid; i < NV4; i += NTHREADS) {
        const int r  = i / (K / 4);
        const int c4 = (i % (K / 4)) * 4;
        const int gr = row0 + r;
        if (gr < nrows) {
#if HAS_ASYNC_LDS
            __builtin_amdgcn_global_load_async_to_lds_b128(
                (__attribute__((address_space(1))) v4i*)(X + (size_t)gr * K + c4),
                (__attribute__((address_space(3))) v4i*)(&As[r * K + c4]),
                /*imm_offset=*/0, /*cpol=*/0);
#else
            const float4 v = *reinterpret_cast<const float4*>(X + (size_t)gr * K + c4);
            As[r * K + c4 + 0] = v.x;
            As[r * K + c4 + 1] = v.y;
            As[r * K + c4 + 2] = v.z;
            As[r * K + c4 + 3] = v.w;
#endif
        } else {
            As[r * K + c4 + 0] = 0.0f;
            As[r * K + c4 + 1] = 0.0f;
            As[r * K + c4 + 2] = 0.0f;
            As[r * K + c4 + 3] = 0.0f;
        }
    }
#if HAS_ASYNC_LDS
    __builtin_amdgcn_s_wait_asynccnt(0);
#endif
    __syncthreads();

    const int lane = tid & 31;
    const int wave = tid >> 5;
    const int col0 = wave * 16;
    const int nn   = lane & 15;        // A: row-in-tile; B: col-in-tile
    const int hk   = (lane >> 4) * 2;  // K sub-pair select

    // Per-lane row scale (A fragment rows: lane holds row nn).
    float srow = 1.0f;
    if (rowscale) {
        const int gr = row0 + nn;
        srow = (gr < nrows) ? rowscale[gr] : 0.0f;
    }

    v8f acc = {0.f, 0.f, 0.f, 0.f, 0.f, 0.f, 0.f, 0.f};

    for (int k0 = 0; k0 < K; k0 += 4) {
        v2f a, b;
        // A 16x4 fragment: lane holds row nn, K = k0+hk, k0+hk+1 (scaled)
        a[0] = As[nn * K + k0 + hk + 0] * srow;
        a[1] = As[nn * K + k0 + hk + 1] * srow;
        // B 4x16 fragment: lane holds col nn, K = k0+hk, k0+hk+1
        b[0] = W[(size_t)(k0 + hk + 0) * OC + col0 + nn];
        b[1] = W[(size_t)(k0 + hk + 1) * OC + col0 + nn];
        acc = __builtin_amdgcn_wmma_f32_16x16x4_f32(
            /*neg_a=*/false, a, /*neg_b=*/false, b,
            /*c_mod=*/(short)0, acc, /*reuse_a=*/false, /*reuse_b=*/false);
    }

    // C/D layout: VGPR j -> row (j + 8*(lane>>4)), col nn
    const int mb = (lane >> 4) * 8;
    if (row0 + 16 <= nrows) {
        // Fast path: full tile, branch-free stores.
        float* yp = Y + (size_t)(row0 + mb) * OC + col0 + nn;
#pragma unroll
        for (int j = 0; j < 8; ++j) {
            yp[(size_t)j * OC] = acc[j];
        }
    } else {
#pragma unroll
        for (int j = 0; j < 8; ++j) {
            const int gr = row0 + mb + j;
            if (gr < nrows) Y[(size_t)gr * OC + col0 + nn] = acc[j];
        }
    }
}

// ---------------- edge scatter: Agg[dst] += M[src] (F = 4 << LOGG floats) ----

__global__ void k_scatter(const float* __restrict__ M, const int* __restrict__ src,
                          const int* __restrict__ dst, float* __restrict__ Agg,
                          int E, int logg) {
    const long long t = (long long)blockIdx.x * blockDim.x + threadIdx.x;
    const int e = (int)(t >> logg);
    if (e >= E) return;
    const int g = (int)(t & ((1 << logg) - 1));
    const int F = 4 << logg;

    const int s = src[e];
    const int d = dst[e];

    // Prefetch a future edge's source row (lowers to global_prefetch_b8).
    const int e2 = e + 256;
    if (g == 0 && e2 < E) {
        __builtin_prefetch(&M[(size_t)src[e2] * F], 0, 1);
    }

    const float4 v = *reinterpret_cast<const float4*>(M + (size_t)s * F + g * 4);
    float* ap = Agg + (size_t)d * F + g * 4;
    atomicAdd(ap + 0, v.x);
    atomicAdd(ap + 1, v.y);
    atomicAdd(ap + 2, v.z);
    atomicAdd(ap + 3, v.w);
}

// ---------------- finalize: out = agg*norm_dst + bias [; relu; *post_scale] --

__global__ void k_fin(const float* __restrict__ agg, const float* __restrict__ ndst,
                      const float* __restrict__ bias, const float* __restrict__ post_scale,
                      float* __restrict__ out, long long total, int logF) {
    const long long i = (long long)blockIdx.x * blockDim.x + threadIdx.x;
    if (i >= total) return;
    const int r = (int)(i >> logF);
    const int c = (int)(i & ((1 << logF) - 1));
    float v = agg[i] * ndst[r] + bias[c];
    if (post_scale) v = fmaxf(v, 0.0f) * post_scale[r];
    out[i] = v;
}

// ---------------- driver ----------------

extern "C" void kernel_launch(void* const* d_in, const int* in_sizes, int n_in,
                              void* d_out, int out_size, void* d_ws, size_t ws_size,
                              hipStream_t stream) {
    const float* x   = (const float*)d_in[0];
    const int*   src = (const int*)d_in[1];
    const int*   dst = (const int*)d_in[2];
    const float* W1  = (const float*)d_in[3];
    const float* b1  = (const float*)d_in[4];
    const float* W2  = (const float*)d_in[5];
    const float* b2  = (const float*)d_in[6];

    const int N = in_sizes[0] / IN_FEATS;  // 50000
    const int E = in_sizes[1];             // 800000

    float* ws       = (float*)d_ws;
    float* norm_src = ws;                                  // [N]
    float* norm_dst = ws + N;                              // [N]
    float* bufA     = ws + 2 * (size_t)N;                  // [N*128] m1 / m2
    float* bufB     = bufA + (size_t)N * HIDDEN;           // [N*128] agg1 / agg2
    float* bufC     = bufB + (size_t)N * HIDDEN;           // [N*128] hs

    const int B = 256;
    auto blocks = [](long long n, int b) { return (int)((n + b - 1) / b); };

    // degrees -> norms
    k_zero<<<blocks(2LL * N, B), B, 0, stream>>>(norm_src, 2LL * N);
    k_zero<<<blocks((long long)N * HIDDEN, B), B, 0, stream>>>(bufB, (long long)N * HIDDEN);
    k_deg<<<blocks(E, B), B, 0, stream>>>(src, dst, norm_src, norm_dst, E);
    k_norm<<<blocks(2LL * N, B), B, 0, stream>>>(norm_src, 2 * N);

    // layer 1: m1 = (x*norm_src) @ W1 ; agg1 = scatter ; hs = relu(agg1*nd+b1)*norm_src
    k_gemm_wmma<IN_FEATS, HIDDEN><<<N / 16 + (N % 16 ? 1 : 0), 256, 0, stream>>>(
        x, W1, norm_src, bufA, N);
    const long long T1 = (long long)E << 5;  // 32 float4-groups per edge
    k_scatter<<<blocks(T1, B), B, 0, stream>>>(bufA, src, dst, bufB, E, 5);
    k_fin<<<blocks((long long)N * HIDDEN, B), B, 0, stream>>>(
        bufB, norm_dst, b1, norm_src, bufC, (long long)N * HIDDEN, 7);

    // layer 2: m2 = hs @ W2 ; agg2 = scatter ; out = agg2*nd + b2
    k_zero<<<blocks((long long)N * OUT_FEATS, B), B, 0, stream>>>(bufB, (long long)N * OUT_FEATS);
    k_gemm_wmma<HIDDEN, OUT_FEATS><<<N / 16 + (N % 16 ? 1 : 0), 128, 0, stream>>>(
        bufC, W2, nullptr, bufA, N);
    const long long T2 = (long long)E << 4;  // 16 float4-groups per edge
    k_scatter<<<blocks(T2, B), B, 0, stream>>>(bufA, src, dst, bufB, E, 4);
    k_fin<<<blocks((long long)N * OUT_FEATS, B), B, 0, stream>>>(
        bufB, norm_dst, b2, nullptr, (float*)d_out, (long long)N * OUT_FEATS, 6);
}